// TransitionDown_90967407329780
// MI455X (gfx1250) — compile-verified
//
#include <hip/hip_runtime.h>
#include <hip/hip_bf16.h>

// ---------------------------------------------------------------------------
// Sizes (fixed by the reference)
// ---------------------------------------------------------------------------
#define NPTS   32768
#define IN_C   64
#define OUT_C  128
#define KNN_K  16
#define MPTS   8192          // NPTS/4
#define BN_EPS 1e-5f

typedef float v2f __attribute__((ext_vector_type(2)));
typedef float v8f __attribute__((ext_vector_type(8)));

// ---------------------------------------------------------------------------
// 1) Farthest-point sampling: one persistent workgroup, coords + min_d in VGPRs
//    1024 threads * 32 points each = 32768 points.
// ---------------------------------------------------------------------------
#define FPS_T 1024
#define PPT   32

__global__ __launch_bounds__(FPS_T) void fps_kernel(const float* __restrict__ p1,
                                                    int* __restrict__ sidx,
                                                    float* __restrict__ p2out) {
    const int tid  = threadIdx.x;
    const int lane = tid & 31;
    const int wave = tid >> 5;

    float px[PPT], py[PPT], pz[PPT], md[PPT];
#pragma unroll
    for (int j = 0; j < PPT; ++j) {
        const int g = tid + j * FPS_T;
        px[j] = p1[g * 3 + 0];
        py[j] = p1[g * 3 + 1];
        pz[j] = p1[g * 3 + 2];
        md[j] = 1.0e10f;
    }

    __shared__ float s_bv[32];
    __shared__ int   s_bi[32];
    __shared__ float s_ptx, s_pty, s_ptz;

    float lx = p1[0], ly = p1[1], lz = p1[2];
    if (tid == 0) {
        sidx[0] = 0;
        p2out[0] = lx; p2out[1] = ly; p2out[2] = lz;
    }

    for (int it = 1; it < MPTS; ++it) {
        float bv = -1.0f;
        int   bi = 0x7fffffff;
#pragma unroll
        for (int j = 0; j < PPT; ++j) {
            const float dx = px[j] - lx, dy = py[j] - ly, dz = pz[j] - lz;
            const float d  = dx * dx + dy * dy + dz * dz;
            md[j] = fminf(md[j], d);
            if (md[j] > bv) { bv = md[j]; bi = tid + j * FPS_T; }
        }
        // wave32 argmax (first-index wins on ties)
#pragma unroll
        for (int off = 16; off > 0; off >>= 1) {
            const float ov = __shfl_xor(bv, off, 32);
            const int   oi = __shfl_xor(bi, off, 32);
            if (ov > bv || (ov == bv && oi < bi)) { bv = ov; bi = oi; }
        }
        if (lane == 0) { s_bv[wave] = bv; s_bi[wave] = bi; }
        __syncthreads();
        if (wave == 0) {
            bv = s_bv[lane]; bi = s_bi[lane];
#pragma unroll
            for (int off = 16; off > 0; off >>= 1) {
                const float ov = __shfl_xor(bv, off, 32);
                const int   oi = __shfl_xor(bi, off, 32);
                if (ov > bv || (ov == bv && oi < bi)) { bv = ov; bi = oi; }
            }
            if (lane == 0) {
                sidx[it] = bi;
                const float qx = p1[bi * 3 + 0];
                const float qy = p1[bi * 3 + 1];
                const float qz = p1[bi * 3 + 2];
                p2out[it * 3 + 0] = qx;
                p2out[it * 3 + 1] = qy;
                p2out[it * 3 + 2] = qz;
                s_ptx = qx; s_pty = qy; s_ptz = qz;
            }
        }
        __syncthreads();
        lx = s_ptx; ly = s_pty; lz = s_ptz;
    }
}

// ---------------------------------------------------------------------------
// 2) h_lin = x @ W via V_WMMA_F32_16X16X4_F32.
//    Block = 256 threads = 8 waves; block owns 16 rows; wave w owns cols
//    [16w, 16w+16). x tile + full W staged in LDS.
// ---------------------------------------------------------------------------
__global__ __launch_bounds__(256) void gemm_kernel(const float* __restrict__ x,
                                                   const float* __restrict__ W,
                                                   float* __restrict__ h) {
    __shared__ float sx[16 * IN_C];       // 4 KB
    __shared__ float sw[IN_C * OUT_C];    // 32 KB

    const int tid  = threadIdx.x;
    const int row0 = blockIdx.x * 16;

    for (int t = tid; t < 16 * IN_C; t += 256) sx[t] = x[row0 * IN_C + t];
    for (int t = tid; t < IN_C * OUT_C; t += 256) sw[t] = W[t];
    __syncthreads();

    const int wave = tid >> 5;
    const int lane = tid & 31;
    const int m    = lane & 15;      // row (A) / col (B) within tile
    const int half = lane >> 4;      // K-half selector
    const int cb   = wave * 16;      // channel base

    v8f acc = {};
#pragma unroll
    for (int k0 = 0; k0 < IN_C; k0 += 4) {
        const int ka = k0 + half * 2;
        v2f a, b;
        a.x = sx[m * IN_C + ka];
        a.y = sx[m * IN_C + ka + 1];
        b.x = sw[ka * OUT_C + cb + m];
        b.y = sw[(ka + 1) * OUT_C + cb + m];
        acc = __builtin_amdgcn_wmma_f32_16x16x4_f32(
            /*neg_a=*/false, a, /*neg_b=*/false, b,
            /*c_mod=*/(short)0, acc, /*reuse_a=*/false, /*reuse_b=*/false);
    }
#pragma unroll
    for (int i = 0; i < 8; ++i)
        h[(row0 + half * 8 + i) * OUT_C + cb + m] = acc[i];
}

// ---------------------------------------------------------------------------
// 3) BatchNorm stats (two-stage, atomic-free) + fused scale/shift, then
//    in-place affine+ReLU.  Note BN(xW+b) == BN(xW): bias is dead.
// ---------------------------------------------------------------------------
__global__ __launch_bounds__(128) void bn_partial(const float* __restrict__ h,
                                                  float* __restrict__ psum,
                                                  float* __restrict__ psq) {
    const int c  = threadIdx.x;          // channel
    const int r0 = blockIdx.x * 128;     // 256 blocks * 128 rows = NPTS
    float s = 0.0f, q = 0.0f;
    for (int r = 0; r < 128; ++r) {
        const float v = h[(r0 + r) * OUT_C + c];
        s += v;
        q += v * v;
    }
    psum[blockIdx.x * OUT_C + c] = s;
    psq [blockIdx.x * OUT_C + c] = q;
}

__global__ __launch_bounds__(128) void bn_final(const float* __restrict__ psum,
                                                const float* __restrict__ psq,
                                                const float* __restrict__ gamma,
                                                const float* __restrict__ beta,
                                                float* __restrict__ ss) {
    const int c = threadIdx.x;
    float s = 0.0f, q = 0.0f;
    for (int b = 0; b < 256; ++b) {
        s += psum[b * OUT_C + c];
        q += psq [b * OUT_C + c];
    }
    const float inv_n = 1.0f / (float)NPTS;
    const float mean  = s * inv_n;
    const float var   = q * inv_n - mean * mean;   // biased var (matches ref)
    const float sc    = gamma[c] * rsqrtf(var + BN_EPS);
    ss[c]          = sc;
    ss[OUT_C + c]  = beta[c] - mean * sc;
}

__global__ __launch_bounds__(256) void bn_apply_relu(float* __restrict__ h,
                                                     const float* __restrict__ ss) {
    const int gid = blockIdx.x * 256 + threadIdx.x;
    const int c   = gid & (OUT_C - 1);
    const float v = h[gid] * ss[c] + ss[OUT_C + c];
    h[gid] = fmaxf(v, 0.0f);
}

// ---------------------------------------------------------------------------
// 4) KNN: one query per thread, p1 tiled through LDS (uniform index -> LDS
//    broadcast), sorted-16 register list with unrolled conditional-swap insert.
//    Top-k order is irrelevant downstream (max-pool), only the set matters.
// ---------------------------------------------------------------------------
#define KNN_TILE 512

__global__ __launch_bounds__(256) void knn_kernel(const float* __restrict__ p1,
                                                  const float* __restrict__ p2,
                                                  int* __restrict__ nbr) {
    __shared__ float sp[KNN_TILE * 3];   // 6 KB
    const int tid = threadIdx.x;
    const int m   = blockIdx.x * 256 + tid;

    const float qx = p2[m * 3 + 0];
    const float qy = p2[m * 3 + 1];
    const float qz = p2[m * 3 + 2];

    float d[KNN_K];
    int   id[KNN_K];
#pragma unroll
    for (int s = 0; s < KNN_K; ++s) { d[s] = 3.0e38f; id[s] = 0; }
    float dmax = 3.0e38f;

    for (int base = 0; base < NPTS; base += KNN_TILE) {
        __syncthreads();
        for (int t = tid; t < KNN_TILE * 3; t += 256) sp[t] = p1[base * 3 + t];
        __syncthreads();
        if (base + KNN_TILE < NPTS)
            __builtin_prefetch(p1 + (base + KNN_TILE) * 3 + tid, 0, 0);

        for (int j = 0; j < KNN_TILE; ++j) {
            const float dx = qx - sp[j * 3 + 0];
            const float dy = qy - sp[j * 3 + 1];
            const float dz = qz - sp[j * 3 + 2];
            const float dd = dx * dx + dy * dy + dz * dz;
            if (dd < dmax) {
                float vv = dd;
                int   vi = base + j;
#pragma unroll
                for (int s = 0; s < KNN_K; ++s) {
                    if (vv < d[s]) {
                        const float tf = d[s]; d[s] = vv; vv = tf;
                        const int   ti = id[s]; id[s] = vi; vi = ti;
                    }
                }
                dmax = d[KNN_K - 1];
            }
        }
    }
#pragma unroll
    for (int s = 0; s < KNN_K; ++s) nbr[m * KNN_K + s] = id[s];
}

// ---------------------------------------------------------------------------
// 5) Gather neighbor features + max-pool over K.
// ---------------------------------------------------------------------------
__global__ __launch_bounds__(OUT_C) void gather_max(const float* __restrict__ h,
                                                    const int* __restrict__ nbr,
                                                    float* __restrict__ y) {
    __shared__ int sn[KNN_K];
    const int m = blockIdx.x;
    const int c = threadIdx.x;
    if (c < KNN_K) sn[c] = nbr[m * KNN_K + c];
    __syncthreads();
    float best = -3.0e38f;
#pragma unroll
    for (int k = 0; k < KNN_K; ++k)
        best = fmaxf(best, h[sn[k] * OUT_C + c]);
    y[m * OUT_C + c] = best;
}

// ---------------------------------------------------------------------------
// Host-side launcher
// ---------------------------------------------------------------------------
extern "C" void kernel_launch(void* const* d_in, const int* in_sizes, int n_in,
                              void* d_out, int out_size, void* d_ws, size_t ws_size,
                              hipStream_t stream) {
    const float* x     = (const float*)d_in[0];   // (N, 64)
    const float* p1    = (const float*)d_in[1];   // (N, 3)
    const float* W     = (const float*)d_in[2];   // (64, 128)
    // d_in[3] = bias: provably dead through BatchNorm (mean subtraction cancels it)
    const float* gamma = (const float*)d_in[4];   // (128,)
    const float* beta  = (const float*)d_in[5];   // (128,)

    float* y  = (float*)d_out;                    // (M, 128)
    float* p2 = (float*)d_out + (size_t)MPTS * OUT_C;  // (M, 3)

    char* ws = (char*)d_ws;
    size_t off = 0;
    int*   sidx = (int*)(ws + off);   off += (size_t)MPTS * sizeof(int);          // 32 KB
    float* h    = (float*)(ws + off); off += (size_t)NPTS * OUT_C * sizeof(float);// 16 MB
    int*   nbr  = (int*)(ws + off);   off += (size_t)MPTS * KNN_K * sizeof(int);  // 512 KB
    float* psum = (float*)(ws + off); off += 256 * OUT_C * sizeof(float);         // 128 KB
    float* psq  = (float*)(ws + off); off += 256 * OUT_C * sizeof(float);         // 128 KB
    float* ss   = (float*)(ws + off); off += 2 * OUT_C * sizeof(float);           // 1 KB
    (void)ws_size; (void)in_sizes; (void)n_in; (void)out_size;

    // Serial-dependency chain on one stream (graph-capture safe).
    fps_kernel   <<<1, FPS_T, 0, stream>>>(p1, sidx, p2);
    gemm_kernel  <<<NPTS / 16, 256, 0, stream>>>(x, W, h);
    bn_partial   <<<256, 128, 0, stream>>>(h, psum, psq);
    bn_final     <<<1, 128, 0, stream>>>(psum, psq, gamma, beta, ss);
    bn_apply_relu<<<(NPTS * OUT_C) / 256, 256, 0, stream>>>(h, ss);
    knn_kernel   <<<MPTS / 256, 256, 0, stream>>>(p1, p2, nbr);
    gather_max   <<<MPTS, OUT_C, 0, stream>>>(h, nbr, y);
}